// GraphQNNGen207_65481071407786
// MI455X (gfx1250) — compile-verified
//
#include <hip/hip_runtime.h>
#include <math.h>

// ---------------------------------------------------------------------------
// GraphQNN pipeline for MI455X (gfx1250, wave32).
//
// Heavy GEMMs (fc1 8.6 GF, fc2 0.54 GF, Gram 0.54 GF) use the CDNA5 full-fp32
// tensor op V_WMMA_F32_16X16X4_F32: exact fp32 semantics (the adjacency matrix
// thresholds squared cosines at 0.8/0.6, so bf16/f16 WMMA would flip
// near-threshold edges), while still running on the matrix pipe.
//
// GEMM kernel is register-blocked 2x4: each wave owns a 32x64 output block
// (8 v8f accumulators = 64 VGPRs). Round-2 used 4x4 (128 acc VGPRs) which,
// combined with the compiler's unroll-2 load pipelining, exceeded 256 VGPRs
// and spilled accumulators to scratch inside the K-loop -- 2x4 keeps total
// pressure ~130 VGPRs (no spills, 2 waves/SIMD) while still reaching ~11
// flops/byte of global traffic with 8 independent WMMAs per K-step for XDL
// latency hiding. Both GEMM operands fit easily in the 192MB L2.
// ---------------------------------------------------------------------------

typedef float v2f __attribute__((ext_vector_type(2)));
typedef float v8f __attribute__((ext_vector_type(8)));

#define THRESHOLD 0.8f
#define SECONDARY 0.6f
#define SEC_W     0.5f
#define BN_EPS    1e-5f

// ---------------------------------------------------------------------------
// conv1: x[1024,1,64,64] -> relu -> maxpool2 -> f1[1024,8,32,32]
// One thread per pooled output. Block = 256 threads => (pw,ph) vary within a
// block, channel c uniform per block => weight loads are scalar (SGPR) loads.
// ---------------------------------------------------------------------------
__global__ void conv1_relu_pool(const float* __restrict__ x,
                                const float* __restrict__ w,
                                const float* __restrict__ bias,
                                float* __restrict__ f1) {
  const int idx = blockIdx.x * blockDim.x + threadIdx.x;
  const int pw = idx & 31, ph = (idx >> 5) & 31, c = (idx >> 10) & 7, b = idx >> 13;
  float wk[9];
#pragma unroll
  for (int i = 0; i < 9; ++i) wk[i] = w[c * 9 + i];
  const float bb = bias[c];
  const float* xp = x + (size_t)b * 4096;  // 64*64
  const int h0 = 2 * ph - 1, w0 = 2 * pw - 1;
  float p[4][4];
#pragma unroll
  for (int i = 0; i < 4; ++i)
#pragma unroll
    for (int j = 0; j < 4; ++j) {
      int hh = h0 + i, ww = w0 + j;
      p[i][j] = (hh >= 0 && hh < 64 && ww >= 0 && ww < 64) ? xp[hh * 64 + ww] : 0.0f;
    }
  float mx = 0.0f;  // relu floor: max(relu(v)) == max(0, convs)
#pragma unroll
  for (int oy = 0; oy < 2; ++oy)
#pragma unroll
    for (int ox = 0; ox < 2; ++ox) {
      float s = bb;
#pragma unroll
      for (int dy = 0; dy < 3; ++dy)
#pragma unroll
        for (int dx = 0; dx < 3; ++dx)
          s = fmaf(p[oy + dy][ox + dx], wk[dy * 3 + dx], s);
      mx = fmaxf(mx, s);
    }
  f1[idx] = mx;  // layout matches [b,c,ph,pw] decomposition of idx
}

// ---------------------------------------------------------------------------
// conv2: f1[1024,8,32,32] -> relu -> maxpool2 -> flat[1024,4096]
// flat[b][c*256 + h*16 + w]  (NCHW reshape order). c uniform per 256-block.
// ---------------------------------------------------------------------------
__global__ void conv2_relu_pool(const float* __restrict__ f1,
                                const float* __restrict__ w,
                                const float* __restrict__ bias,
                                float* __restrict__ flat) {
  const int idx = blockIdx.x * blockDim.x + threadIdx.x;
  const int pw = idx & 15, ph = (idx >> 4) & 15, c = (idx >> 8) & 15, b = idx >> 12;
  const float bb = bias[c];
  float acc[2][2] = {{bb, bb}, {bb, bb}};
  const int h0 = 2 * ph - 1, w0 = 2 * pw - 1;
  for (int ic = 0; ic < 8; ++ic) {
    const float* xp = f1 + ((size_t)b * 8 + ic) * 1024;  // 32*32
    const float* wk = w + ((size_t)c * 8 + ic) * 9;
    float p[4][4];
#pragma unroll
    for (int i = 0; i < 4; ++i)
#pragma unroll
      for (int j = 0; j < 4; ++j) {
        int hh = h0 + i, ww = w0 + j;
        p[i][j] = (hh >= 0 && hh < 32 && ww >= 0 && ww < 32) ? xp[hh * 32 + ww] : 0.0f;
      }
#pragma unroll
    for (int oy = 0; oy < 2; ++oy)
#pragma unroll
      for (int ox = 0; ox < 2; ++ox)
#pragma unroll
        for (int dy = 0; dy < 3; ++dy)
#pragma unroll
          for (int dx = 0; dx < 3; ++dx)
            acc[oy][ox] = fmaf(p[oy + dy][ox + dx], wk[dy * 3 + dx], acc[oy][ox]);
  }
  float mx = fmaxf(fmaxf(acc[0][0], acc[0][1]), fmaxf(acc[1][0], acc[1][1]));
  flat[idx] = fmaxf(mx, 0.0f);
}

// ---------------------------------------------------------------------------
// Register-blocked fp32 WMMA GEMM:  out = epi(A @ W^T [+ bias])
//   A: [M,K] row-major, W: [N,K] row-major, K % 4 == 0, M,N % 128 == 0.
// Block = 256 threads = 8 waves arranged 4(m) x 2(n); block covers 128x128.
// Each wave owns a 32x64 sub-block = 2x4 grid of 16x16 WMMA tiles
// (8 v8f accumulators). Per K-step: 2 A-frags + 4 B-frags feed 8 independent
// v_wmma_f32_16x16x4_f32.
// Fragment layouts per CDNA5 ISA 7.12.2:
//   A 16x4 : lanes 0-15 hold K=0,1 (VGPR0,1); lanes 16-31 hold K=2,3.
//   B 4x16 : mirrored (rows K split across lane halves, cols across lanes).
//   C/D    : VGPR r -> row r (lanes 0-15) / row r+8 (lanes 16-31).
// EPI=0: tanh(x + bias[col]).  EPI=1: fid=x^2 threshold + zero diagonal.
// ---------------------------------------------------------------------------
template <int EPI>
__global__ __launch_bounds__(256)
void gemm_wmma_f32(const float* __restrict__ A,
                   const float* __restrict__ W,
                   const float* __restrict__ bias,
                   float* __restrict__ out,
                   int M, int N, int K) {
  const int lane = threadIdx.x & 31;
  const int wv   = threadIdx.x >> 5;            // 0..7
  const int m0   = blockIdx.y * 128 + (wv >> 1) * 32;   // 4 waves along M
  const int n0   = blockIdx.x * 128 + (wv & 1) * 64;    // 2 waves along N
  const int half = lane >> 4;                   // 0: K pair (0,1); 1: (2,3)
  const int l    = lane & 15;

  // Per-lane row pointers for the 2 A-tiles and 4 B-tiles (offset by K half).
  const float* arow[2];
  const float* wrow[4];
#pragma unroll
  for (int i = 0; i < 2; ++i) arow[i] = A + (size_t)(m0 + 16 * i + l) * K + 2 * half;
#pragma unroll
  for (int j = 0; j < 4; ++j) wrow[j] = W + (size_t)(n0 + 16 * j + l) * K + 2 * half;

  v8f acc[2][4] = {};
  for (int k0 = 0; k0 < K; k0 += 4) {
    v2f a[2], b[4];
#pragma unroll
    for (int i = 0; i < 2; ++i) a[i] = *reinterpret_cast<const v2f*>(arow[i] + k0);
#pragma unroll
    for (int j = 0; j < 4; ++j) b[j] = *reinterpret_cast<const v2f*>(wrow[j] + k0);
#pragma unroll
    for (int i = 0; i < 2; ++i)
#pragma unroll
      for (int j = 0; j < 4; ++j)
        acc[i][j] = __builtin_amdgcn_wmma_f32_16x16x4_f32(
            /*neg_a=*/false, a[i], /*neg_b=*/false, b[j],
            /*c_mod=*/(short)0, acc[i][j], /*reuse_a=*/false, /*reuse_b=*/false);
  }

  // Epilogue: C/D VGPR r -> row (tile_m + r + 8*half), col (tile_n + l).
#pragma unroll
  for (int i = 0; i < 2; ++i)
#pragma unroll
    for (int j = 0; j < 4; ++j) {
      const int col = n0 + 16 * j + l;
      const float bval = (EPI == 0) ? bias[col] : 0.0f;
#pragma unroll
      for (int r = 0; r < 8; ++r) {
        const int row = m0 + 16 * i + r + 8 * half;
        float v = acc[i][j][r];
        if (EPI == 0) {
          v = tanhf(v + bval);
        } else {
          const float fid = v * v;
          v = (fid >= THRESHOLD) ? 1.0f : ((fid >= SECONDARY) ? SEC_W : 0.0f);
          if (row == col) v = 0.0f;  // no self-edges
        }
        out[(size_t)row * N + col] = v;
      }
    }
}

// ---------------------------------------------------------------------------
// Row L2-normalize: nrm[r,:] = last[r,:] / (||last[r,:]|| + 1e-12).
// One wave per row (256 cols, 8 per lane), wave32 xor-shuffle reduction.
// ---------------------------------------------------------------------------
__global__ void rownorm(const float* __restrict__ last, float* __restrict__ nrm) {
  const int wv = threadIdx.x >> 5, lane = threadIdx.x & 31;
  const int row = blockIdx.x * 8 + wv;
  const float* p = last + (size_t)row * 256;
  float s = 0.0f;
#pragma unroll
  for (int i = lane; i < 256; i += 32) { const float v = p[i]; s = fmaf(v, v, s); }
#pragma unroll
  for (int off = 16; off > 0; off >>= 1) s += __shfl_xor(s, off, 32);
  const float inv = 1.0f / (sqrtf(s) + 1e-12f);
  float* q = nrm + (size_t)row * 256;
#pragma unroll
  for (int i = lane; i < 256; i += 32) q[i] = p[i] * inv;
}

// ---------------------------------------------------------------------------
// BatchNorm1d training-mode stats over the batch axis (1024 rows, 256 cols).
// One thread per column; reads are coalesced across the 256-thread block.
// ---------------------------------------------------------------------------
__global__ void bn_stats(const float* __restrict__ last,
                         float* __restrict__ mean_out,
                         float* __restrict__ invstd_out) {
  const int c = threadIdx.x;  // single block of 256
  float s = 0.0f, s2 = 0.0f;
  for (int r = 0; r < 1024; ++r) {
    const float v = last[(size_t)r * 256 + c];
    s += v;
    s2 = fmaf(v, v, s2);
  }
  const float mean = s * (1.0f / 1024.0f);
  const float var  = fmaxf(s2 * (1.0f / 1024.0f) - mean * mean, 0.0f);  // biased
  mean_out[c]   = mean;
  invstd_out[c] = rsqrtf(var + BN_EPS);
}

__global__ void bn_apply(const float* __restrict__ last,
                         const float* __restrict__ mean,
                         const float* __restrict__ invstd,
                         const float* __restrict__ gamma,
                         const float* __restrict__ beta,
                         float* __restrict__ out) {
  const int idx = blockIdx.x * blockDim.x + threadIdx.x;  // 1024*256 threads
  const int c = idx & 255;
  out[idx] = fmaf((last[idx] - mean[c]) * invstd[c], gamma[c], beta[c]);
}

// ---------------------------------------------------------------------------
// Host launcher. Inputs (setup_inputs order):
//  0:x 1:cw1 2:cb1 3:cw2 4:cb2 5:w1 6:b1 7:w2 8:b2 9:gamma 10:beta
// d_out = [ out(1024*256) | adj(1024*1024) ] fp32.
// ---------------------------------------------------------------------------
extern "C" void kernel_launch(void* const* d_in, const int* in_sizes, int n_in,
                              void* d_out, int out_size, void* d_ws, size_t ws_size,
                              hipStream_t stream) {
  const float* x     = (const float*)d_in[0];
  const float* cw1   = (const float*)d_in[1];
  const float* cb1   = (const float*)d_in[2];
  const float* cw2   = (const float*)d_in[3];
  const float* cb2   = (const float*)d_in[4];
  const float* w1    = (const float*)d_in[5];
  const float* b1    = (const float*)d_in[6];
  const float* w2    = (const float*)d_in[7];
  const float* b2    = (const float*)d_in[8];
  const float* gamma = (const float*)d_in[9];
  const float* beta  = (const float*)d_in[10];

  float* outp = (float*)d_out;             // [1024,256] BN output
  float* adj  = outp + 1024 * 256;         // [1024,1024] adjacency

  // Workspace carve-up (floats): ~56.6 MB total.
  float* ws     = (float*)d_ws;
  float* f1     = ws;                       // 1024*8*32*32   = 8388608
  float* flat   = f1 + 8388608;             // 1024*4096      = 4194304
  float* h1     = flat + 4194304;           // 1024*1024      = 1048576
  float* last   = h1 + 1048576;             // 1024*256       = 262144
  float* nrm    = last + 262144;            // 1024*256       = 262144
  float* mean   = nrm + 262144;             // 256
  float* invstd = mean + 256;               // 256

  // 1) CNN feature extractor
  conv1_relu_pool<<<8388608 / 256, 256, 0, stream>>>(x, cw1, cb1, f1);
  conv2_relu_pool<<<4194304 / 256, 256, 0, stream>>>(f1, cw2, cb2, flat);

  // 2) fc1: h1 = tanh(flat @ w1^T + b1)   [1024,1024], K=4096
  gemm_wmma_f32<0><<<dim3(1024 / 128, 1024 / 128), 256, 0, stream>>>(
      flat, w1, b1, h1, 1024, 1024, 4096);

  // 3) fc2: last = tanh(h1 @ w2^T + b2)   [1024,256], K=1024
  gemm_wmma_f32<0><<<dim3(256 / 128, 1024 / 128), 256, 0, stream>>>(
      h1, w2, b2, last, 1024, 256, 1024);

  // 4) fidelity adjacency: nrm = normalize(last); adj = thr((nrm@nrm^T)^2)
  rownorm<<<1024 / 8, 256, 0, stream>>>(last, nrm);
  gemm_wmma_f32<1><<<dim3(1024 / 128, 1024 / 128), 256, 0, stream>>>(
      nrm, nrm, nullptr, adj, 1024, 1024, 256);

  // 5) BatchNorm1d (training mode, biased variance)
  bn_stats<<<1, 256, 0, stream>>>(last, mean, invstd);
  bn_apply<<<(1024 * 256) / 256, 256, 0, stream>>>(last, mean, invstd, gamma, beta, outp);
}